// Harmonic_41944650613267
// MI455X (gfx1250) — compile-verified
//
#include <hip/hip_runtime.h>
#include <math.h>

typedef __attribute__((ext_vector_type(2))) float v2f;
typedef __attribute__((ext_vector_type(4))) float v4f;
typedef __attribute__((ext_vector_type(8))) float v8f;

#define NB   8
#define TF   400
#define NF   512
#define NH   128
#define NT   (TF*256)          /* 102400 */
#define SRF  16000.0f
#define LOG10_C 2.302585092994046f
#define RPOS (399.0f/102399.0f)
#define CHUNK 1024
#define NCH  (NT/CHUNK)        /* 100 chunks per batch */

// ---------------------------------------------------------------------------
// K1: alphas_small[b,tf,h] = mod_sigmoid(x @ Wa^T + ba)
// fp32 WMMA 16x16x4: one wave per 16x16 tile, K-loop over F=512.
// ---------------------------------------------------------------------------
__global__ __launch_bounds__(32)
void k_gemm_alphas(const float* __restrict__ x, const float* __restrict__ Wa,
                   const float* __restrict__ ba, float* __restrict__ Asmall) {
  const int mt = blockIdx.x;            // 0..199
  const int nt = blockIdx.y;            // 0..7
  const int lane = threadIdx.x;
  const int g = lane >> 4;
  const int l = lane & 15;
  const float* xrow = x  + (size_t)(mt*16 + l) * NF;
  const float* wrow = Wa + (size_t)(nt*16 + l) * NF;

  v8f c = {0.f,0.f,0.f,0.f,0.f,0.f,0.f,0.f};
#pragma unroll 4
  for (int k = 0; k < NF; k += 4) {
    v2f a = *(const v2f*)(xrow + k + 2*g);
    v2f b = *(const v2f*)(wrow + k + 2*g);
    c = __builtin_amdgcn_wmma_f32_16x16x4_f32(false, a, false, b,
                                              (short)0, c, false, false);
  }

  const int col = nt*16 + l;
  const float bias = ba[col];
#pragma unroll
  for (int j = 0; j < 8; ++j) {
    const int row = mt*16 + j + 8*g;
    float v = c[j] + bias;
    float s = __builtin_amdgcn_rcpf(1.0f + __expf(-v));   // sigmoid (v_rcp)
    float p = __expf(LOG10_C * __logf(s));                // s^ln(10)
    Asmall[(size_t)row*NH + col] = 2.0f*p + 1e-7f;        // mod_sigmoid
  }
}

// ---------------------------------------------------------------------------
// K2: amp_small = softplus(x @ Wamp^T + bamp). One wave per row.
// ---------------------------------------------------------------------------
__global__ __launch_bounds__(256)
void k_amp(const float* __restrict__ x, const float* __restrict__ Wamp,
           const float* __restrict__ bamp, float* __restrict__ ampSmall) {
  const int wave = threadIdx.x >> 5;
  const int lane = threadIdx.x & 31;
  const int row  = blockIdx.x * 8 + wave;            // 0..3199
  const float* xr = x + (size_t)row * NF;
  float s = 0.f;
  for (int k = lane; k < NF; k += 32) s += xr[k] * Wamp[k];
  for (int off = 16; off; off >>= 1) s += __shfl_xor(s, off, 32);
  if (lane == 0) {
    float v = s + bamp[0];
    ampSmall[row] = (v > 20.f) ? v : log1pf(__expf(v));
  }
}

// ---------------------------------------------------------------------------
// K2b: per-frame row sums of alphas_small. Normalizer of a lerped signal is
// the lerp of row sums, so this kills the denominator pass in K5.
// ---------------------------------------------------------------------------
__global__ __launch_bounds__(256)
void k_rowsum(const float* __restrict__ Asmall, float* __restrict__ rowSum) {
  const int wave = threadIdx.x >> 5;
  const int lane = threadIdx.x & 31;
  const int row  = blockIdx.x * 8 + wave;
  const float* r = Asmall + (size_t)row * NH;
  float s = r[lane] + r[lane+32] + r[lane+64] + r[lane+96];
  for (int off = 16; off; off >>= 1) s += __shfl_xor(s, off, 32);
  if (lane == 0) rowSum[row] = s;
}

// f0u (normalized) at output sample t for batch b
__device__ __forceinline__ float f0u_at(const float* f0, int b, int t) {
  float pos = (float)t * RPOS;
  int i0 = (int)pos; if (i0 > TF-1) i0 = TF-1;
  int i1 = min(i0 + 1, TF-1);
  float w = pos - (float)i0;
  const float* fb = f0 + b*TF;
  return (fb[i0] + (fb[i1] - fb[i0]) * w) * (1.0f/SRF);
}

// ---------------------------------------------------------------------------
// K3: per-1024-sample chunk sums of f0u
// ---------------------------------------------------------------------------
__global__ __launch_bounds__(256)
void k_chunksum(const float* __restrict__ f0, float* __restrict__ chunkSum) {
  const int b = blockIdx.y, cch = blockIdx.x;        // NCH chunks/batch
  float s = 0.f;
#pragma unroll
  for (int j = 0; j < 4; ++j)
    s += f0u_at(f0, b, cch*CHUNK + threadIdx.x*4 + j);
  __shared__ float sm[256];
  sm[threadIdx.x] = s;
  __syncthreads();
  for (int off = 128; off; off >>= 1) {
    if (threadIdx.x < off) sm[threadIdx.x] += sm[threadIdx.x + off];
    __syncthreads();
  }
  if (threadIdx.x == 0) chunkSum[b*NCH + cch] = sm[0];
}

// ---------------------------------------------------------------------------
// K4: exclusive scan of chunk sums per batch, reduced mod 1 (revolutions).
// ---------------------------------------------------------------------------
__global__ void k_scanchunks(const float* __restrict__ chunkSum,
                             float* __restrict__ chunkBase) {
  const int b = blockIdx.x;
  if (threadIdx.x == 0) {
    float run = 0.f;
    for (int cch = 0; cch < NCH; ++cch) {
      chunkBase[b*NCH + cch] = run;
      run += chunkSum[b*NCH + cch];
      run -= floorf(run);
    }
  }
}

// ---------------------------------------------------------------------------
// K5: streaming pass. 256 threads x 4 consecutive t each = 1024 t per block.
//  - per-h output is a b128 NT store (4 t per lane, coalesced 512B/wave)
//  - 6 alpha frames (3KB) cached in LDS; denom via row-sum lerp
//  - phase completed by in-block Hillis-Steele scan, mod-1 reduced
// ---------------------------------------------------------------------------
__global__ __launch_bounds__(256)
void k_main(const float* __restrict__ f0, const float* __restrict__ Asmall,
            const float* __restrict__ ampSmall, const float* __restrict__ rowSum,
            const float* __restrict__ chunkBase,
            float* __restrict__ outP, float* __restrict__ ampP,
            float* __restrict__ alphaP) {
  const int b = blockIdx.y, cch = blockIdx.x;
  const int tid = threadIdx.x;
  const int t0 = cch*CHUNK + tid*4;

  __shared__ float scan[256];
  __shared__ float frames[6*NH];
  __shared__ float ff0[6];
  __shared__ float famp[6];
  __shared__ float fsum[6];

  int fbase = (int)((float)(cch*CHUNK) * RPOS);
  if (fbase > TF-1) fbase = TF-1;

  for (int j = tid; j < 6*NH; j += 256) {
    int fr = min(fbase + j/NH, TF-1);
    frames[j] = Asmall[(size_t)(b*TF + fr)*NH + (j & (NH-1))];
  }
  if (tid < 6) {
    int fr = min(fbase + tid, TF-1);
    ff0[tid]  = f0[b*TF + fr];
    famp[tid] = ampSmall[b*TF + fr];
    fsum[tid] = rowSum[b*TF + fr];
  }
  __syncthreads();

  int d0[4], d1[4];
  float w[4], f0n[4], pre[4];
  float lsum = 0.f;
#pragma unroll
  for (int j = 0; j < 4; ++j) {
    int t = t0 + j;
    float pos = (float)t * RPOS;
    int i0 = (int)pos; if (i0 > TF-1) i0 = TF-1;
    int i1 = min(i0 + 1, TF-1);
    w[j]  = pos - (float)i0;
    d0[j] = max(0, min(5, i0 - fbase));
    d1[j] = max(0, min(5, i1 - fbase));
    f0n[j] = (ff0[d0[j]] + (ff0[d1[j]] - ff0[d0[j]]) * w[j]) * (1.0f/SRF);
    pre[j] = lsum;                       // exclusive within thread
    lsum += f0n[j];
  }

  // inclusive scan of per-thread sums across 256 threads
  float val = lsum;
  scan[tid] = val;
  __syncthreads();
  for (int off = 1; off < 256; off <<= 1) {
    float add = (tid >= off) ? scan[tid - off] : 0.f;
    __syncthreads();
    val += add;
    scan[tid] = val;
    __syncthreads();
  }
  float excl = val - lsum;               // exclusive prefix of this thread
  float base = chunkBase[b*NCH + cch];

  float m[4], ampv[4], inv[4];
#pragma unroll
  for (int j = 0; j < 4; ++j) {
    float mm = base + excl + pre[j] + f0n[j];   // inclusive cumsum, cycles
    m[j] = mm - floorf(mm);
    ampv[j] = famp[d0[j]] + (famp[d1[j]] - famp[d0[j]]) * w[j];
    float den = fsum[d0[j]] + (fsum[d1[j]] - fsum[d0[j]]) * w[j];
    inv[j] = __builtin_amdgcn_rcpf(den);
  }

  float acc[4] = {0.f,0.f,0.f,0.f};
  float ph[4]  = {0.f,0.f,0.f,0.f};
#pragma unroll 2
  for (int h = 0; h < NH; ++h) {
    v4f vec;
    float hh = (float)(h + 1);
#pragma unroll
    for (int j = 0; j < 4; ++j) {
      float a0 = frames[d0[j]*NH + h];
      float a1 = frames[d1[j]*NH + h];
      float al = (a0 + (a1 - a0) * w[j]) * inv[j];
      vec[j] = al;
      ph[j] += m[j];                              // ph = m*(h+1)
      if (f0n[j] * hh < 0.5f) {                   // antialias mask
        float arg = __builtin_amdgcn_fractf(ph[j]);
        acc[j] += __builtin_amdgcn_sinf(arg) * al; // v_sin_f32: sin(2pi*x)
      }
    }
    __builtin_nontemporal_store(vec, (v4f*)(alphaP + (size_t)(b*NH + h)*NT + t0));
  }

  v4f o, a;
#pragma unroll
  for (int j = 0; j < 4; ++j) { o[j] = acc[j] * ampv[j]; a[j] = ampv[j]; }
  *(v4f*)(outP + (size_t)b*NT + t0) = o;
  *(v4f*)(ampP + (size_t)b*NT + t0) = a;
}

// ---------------------------------------------------------------------------
extern "C" void kernel_launch(void* const* d_in, const int* in_sizes, int n_in,
                              void* d_out, int out_size, void* d_ws, size_t ws_size,
                              hipStream_t stream) {
  const float* x    = (const float*)d_in[0];   // (8,400,512)
  const float* f0   = (const float*)d_in[1];   // (8,400)
  const float* Wa   = (const float*)d_in[2];   // (128,512)
  const float* ba   = (const float*)d_in[3];   // (128,)
  const float* Wamp = (const float*)d_in[4];   // (1,512)
  const float* bamp = (const float*)d_in[5];   // (1,)

  float* outP   = (float*)d_out;                         // (8,1,102400)
  float* ampP   = outP + (size_t)NB*NT;                  // (8,1,102400)
  float* alphaP = ampP + (size_t)NB*NT;                  // (8,128,102400)

  float* Asmall    = (float*)d_ws;                       // 3200*128
  float* ampSmall  = Asmall + (size_t)NB*TF*NH;          // 3200
  float* rowSum    = ampSmall + NB*TF;                   // 3200
  float* chunkSum  = rowSum + NB*TF;                     // 800
  float* chunkBase = chunkSum + NB*NCH;                  // 800

  dim3 g1(200, 8);
  k_gemm_alphas<<<g1, 32, 0, stream>>>(x, Wa, ba, Asmall);
  k_amp<<<400, 256, 0, stream>>>(x, Wamp, bamp, ampSmall);
  k_rowsum<<<400, 256, 0, stream>>>(Asmall, rowSum);
  dim3 g3(NCH, 8);
  k_chunksum<<<g3, 256, 0, stream>>>(f0, chunkSum);
  k_scanchunks<<<8, 32, 0, stream>>>(chunkSum, chunkBase);
  dim3 g5(NCH, 8);
  k_main<<<g5, 256, 0, stream>>>(f0, Asmall, ampSmall, rowSum, chunkBase,
                                 outP, ampP, alphaP);
}